// GIN_drug_23845658428385
// MI455X (gfx1250) — compile-verified
//
#include <hip/hip_runtime.h>

// ---------------- problem constants (match reference) ----------------
#define NN      40000     // nodes
#define EE      640000    // edges
#define IN_DIM  75        // atom feature dim
#define H0S     80        // padded stride for layer-0 h (75 -> 80)
#define H0K     76        // padded K for layer-0 GEMM1 (mult of 4)
#define HDIM    128       // hidden dim
#define GG      1024      // graphs
#define POOLW   (3*HDIM)  // 384 concat width
#define BN_EPS  1e-5f

typedef float v2f __attribute__((ext_vector_type(2)));
typedef float v8f __attribute__((ext_vector_type(8)));

// order-preserving float <-> uint mapping for atomic max pooling
__device__ __forceinline__ unsigned fflip(float f) {
    unsigned u = __float_as_uint(f);
    return (u & 0x80000000u) ? ~u : (u | 0x80000000u);
}
__device__ __forceinline__ float funflip(unsigned u) {
    return (u & 0x80000000u) ? __uint_as_float(u & 0x7fffffffu)
                             : __uint_as_float(~u);
}

// ---------------- h = x (with zero pad columns) ----------------
__global__ void k_pad_copy(const float* __restrict__ x, float* __restrict__ h,
                           int n, int dim, int hs) {
    int t = blockIdx.x * blockDim.x + threadIdx.x;
    int total = n * hs;
    if (t >= total) return;
    int row = t / hs, c = t - row * hs;
    h[t] = (c < dim) ? x[(size_t)row * dim + c] : 0.0f;
}

// ------- weight prep: transpose [din x 128] -> padded [128 x kp], zero pad rows -------
__global__ void k_wt_pad(const float* __restrict__ w, float* __restrict__ wt,
                         int din, int kp) {
    int t = blockIdx.x * blockDim.x + threadIdx.x;
    int total = HDIM * kp;
    if (t >= total) return;
    int n = t / kp, k = t - n * kp;
    wt[t] = (k < din) ? w[(size_t)k * HDIM + n] : 0.0f;
}

// ---------------- h[dst] += x[src]  (one wave per edge) ----------------
__global__ void k_scatter(const float* __restrict__ x, float* __restrict__ h,
                          const long long* __restrict__ adj, int nedges,
                          int dim, int hs) {
    int w = (blockIdx.x * blockDim.x + threadIdx.x) >> 5;
    int lane = threadIdx.x & 31;
    if (w >= nedges) return;
    long long s = adj[w];            // src row of (2,E) adjacency
    long long d = adj[nedges + w];   // dst row
    const float* xs = x + (size_t)s * dim;
    float* hd = h + (size_t)d * hs;
    for (int k = lane; k < dim; k += 32)
        atomicAdd(&hd[k], xs[k]);    // global_atomic_add_f32 (L2 RMW)
}

// ---------------- pool buffer init / finalize ----------------
__global__ void k_pool_init(unsigned* __restrict__ p, int n) {
    int t = blockIdx.x * blockDim.x + threadIdx.x;
    if (t < n) p[t] = 0x007FFFFFu;   // fflip(-inf)
}
__global__ void k_pool_fin(const unsigned* __restrict__ p, float* __restrict__ o, int n) {
    int t = blockIdx.x * blockDim.x + threadIdx.x;
    if (t < n) o[t] = funflip(p[t]);
}

// ---- fused MLP: relu(h@W1+b1)@W2+b2 -> relu -> BN -> out + maxpool ----
// One wave per 16-row tile; 4 waves / block.
// Weights pre-transposed+padded: wt1 [128 x KP], wt2 [128 x 128]; a lane's
// B-fragment (W[kk][n], W[kk+1][n]) is one contiguous b64 load -> branchless.
// 8 column-tile accumulators in registers (8 independent WMMA chains);
// accumulators start at 0 (inline SRC2=0 on first WMMA), bias added in epilogue.
template<int HS, int KP>
__global__ __launch_bounds__(128) void k_mlp(
    const float* __restrict__ h,
    const float* __restrict__ wt1, const float* __restrict__ b1,
    const float* __restrict__ wt2, const float* __restrict__ b2,
    const float* __restrict__ gamma, const float* __restrict__ beta,
    const float* __restrict__ mean,  const float* __restrict__ var,
    float* __restrict__ xout,
    unsigned* __restrict__ pool, const long long* __restrict__ ibatch,
    int pool_off, int n_nodes)
{
    __shared__ __align__(16) float t_lds[4][16 * 132]; // per-wave 16x128 (+pad)

    const int lane = threadIdx.x & 31;
    const int wave = threadIdx.x >> 5;
    const int grp  = lane >> 4;      // 0: K={k,k+1}, 1: K={k+2,k+3}
    const int r    = lane & 15;
    const int tile = blockIdx.x * 4 + wave;
    const int row0 = tile * 16;
    if (row0 >= n_nodes) return;     // wave-uniform: EXEC stays all-ones

    float* tl = &t_lds[wave][0];

    // ---------- GEMM1: T = relu(h @ W1 + b1), T staged to LDS ----------
    {
        v8f acc[8];
#pragma unroll
        for (int ct = 0; ct < 8; ++ct)
            acc[ct] = (v8f){0.f, 0.f, 0.f, 0.f, 0.f, 0.f, 0.f, 0.f};
        const float* arow = h + (size_t)(row0 + r) * HS + 2 * grp;
#pragma unroll 4
        for (int k0 = 0; k0 < KP; k0 += 4) {
            v2f a = *(const v2f*)(arow + k0);
#pragma unroll
            for (int ct = 0; ct < 8; ++ct) {
                v2f b = *(const v2f*)(wt1 + (size_t)(ct * 16 + r) * KP + k0 + 2 * grp);
                acc[ct] = __builtin_amdgcn_wmma_f32_16x16x4_f32(
                              false, a, false, b, (short)0, acc[ct], false, false);
            }
        }
#pragma unroll
        for (int ct = 0; ct < 8; ++ct) {
            const int n = ct * 16 + r;
            const float bias = b1[n];
#pragma unroll
            for (int v = 0; v < 8; ++v) {
                float t = acc[ct][v] + bias;                       // bias
                t = t > 0.0f ? t : 0.0f;                           // ReLU
                tl[(v + 8 * grp) * 132 + n] = t;                   // C-layout -> LDS
            }
        }
    }
    asm volatile("s_wait_dscnt 0x0" ::: "memory");       // LDS writes visible

    // graph ids for this tile's rows (fixed across col tiles)
    int gid[8];
#pragma unroll
    for (int v = 0; v < 8; ++v)
        gid[v] = (int)ibatch[row0 + v + 8 * grp];

    // ---------- GEMM2: O = T @ W2 + b2 -> relu -> BN -> store + maxpool ----------
    {
        v8f acc[8];
#pragma unroll
        for (int ct = 0; ct < 8; ++ct)
            acc[ct] = (v8f){0.f, 0.f, 0.f, 0.f, 0.f, 0.f, 0.f, 0.f};
        const float* arow = tl + r * 132 + 2 * grp;
#pragma unroll 4
        for (int k0 = 0; k0 < HDIM; k0 += 4) {
            v2f a = *(const v2f*)(arow + k0);                      // ds_load_b64
#pragma unroll
            for (int ct = 0; ct < 8; ++ct) {
                v2f b = *(const v2f*)(wt2 + (size_t)(ct * 16 + r) * HDIM + k0 + 2 * grp);
                acc[ct] = __builtin_amdgcn_wmma_f32_16x16x4_f32(
                              false, a, false, b, (short)0, acc[ct], false, false);
            }
        }
#pragma unroll
        for (int ct = 0; ct < 8; ++ct) {
            const int n = ct * 16 + r;
            const float bias = b2[n];
            // eval-mode BN folded to per-column scale/shift (lane owns column n)
            const float sc = gamma[n] * rsqrtf(var[n] + BN_EPS);
            const float sh = beta[n] - mean[n] * sc;
#pragma unroll
            for (int v = 0; v < 8; ++v) {
                const int row = row0 + v + 8 * grp;
                float o = acc[ct][v] + bias;                       // bias
                o = o > 0.0f ? o : 0.0f;                           // ReLU
                o = o * sc + sh;                                   // BatchNorm
                xout[(size_t)row * HDIM + n] = o;                  // next layer input
                atomicMax(&pool[(size_t)gid[v] * POOLW + pool_off + n], fflip(o));
            }
        }
    }
}

// ---------------- launcher ----------------
extern "C" void kernel_launch(void* const* d_in, const int* in_sizes, int n_in,
                              void* d_out, int out_size, void* d_ws, size_t ws_size,
                              hipStream_t stream) {
    (void)in_sizes; (void)n_in; (void)out_size; (void)ws_size;

    const float*     feat = (const float*)d_in[0];
    const long long* adj  = (const long long*)d_in[1];   // int64 (2,E)
    const long long* ib   = (const long long*)d_in[2];   // int64 (N,)
    const float* w1[3] = {(const float*)d_in[3],  (const float*)d_in[4],  (const float*)d_in[5]};
    const float* b1[3] = {(const float*)d_in[6],  (const float*)d_in[7],  (const float*)d_in[8]};
    const float* w2[3] = {(const float*)d_in[9],  (const float*)d_in[10], (const float*)d_in[11]};
    const float* b2[3] = {(const float*)d_in[12], (const float*)d_in[13], (const float*)d_in[14]};
    const float* ga[3] = {(const float*)d_in[15], (const float*)d_in[16], (const float*)d_in[17]};
    const float* be[3] = {(const float*)d_in[18], (const float*)d_in[19], (const float*)d_in[20]};
    const float* mu[3] = {(const float*)d_in[21], (const float*)d_in[22], (const float*)d_in[23]};
    const float* va[3] = {(const float*)d_in[24], (const float*)d_in[25], (const float*)d_in[26]};

    // workspace layout (~77 MB, all re-initialized every call)
    float* h0 = (float*)d_ws;                      // NN * 80   padded layer-0 h
    float* xA = h0 + (size_t)NN * H0S;             // NN * 128
    float* hB = xA + (size_t)NN * HDIM;            // NN * 128  aggr buffer
    float* xC = hB + (size_t)NN * HDIM;            // NN * 128
    unsigned* pool = (unsigned*)(xC + (size_t)NN * HDIM);  // GG * 384 u32
    float* wt1_0 = (float*)(pool + (size_t)GG * POOLW);    // 128 * 76
    float* wt1_1 = wt1_0 + (size_t)HDIM * H0K;             // 128 * 128
    float* wt1_2 = wt1_1 + (size_t)HDIM * HDIM;
    float* wt2_0 = wt1_2 + (size_t)HDIM * HDIM;
    float* wt2_1 = wt2_0 + (size_t)HDIM * HDIM;
    float* wt2_2 = wt2_1 + (size_t)HDIM * HDIM;

    const int T256 = 256;
    const int mlp_blocks = (NN / 16) / 4;          // 2500 tiles / 4 waves = 625
    const int sc_blocks  = EE / 8;                 // 1 wave per edge, 8 waves/block

    k_pool_init<<<(GG * POOLW + T256 - 1) / T256, T256, 0, stream>>>(pool, GG * POOLW);

    // weight prep: transpose + zero-pad (tiny; keeps GEMM inner loops branchless)
    k_wt_pad<<<(HDIM * H0K  + T256 - 1) / T256, T256, 0, stream>>>(w1[0], wt1_0, IN_DIM, H0K);
    k_wt_pad<<<(HDIM * HDIM + T256 - 1) / T256, T256, 0, stream>>>(w1[1], wt1_1, HDIM, HDIM);
    k_wt_pad<<<(HDIM * HDIM + T256 - 1) / T256, T256, 0, stream>>>(w1[2], wt1_2, HDIM, HDIM);
    k_wt_pad<<<(HDIM * HDIM + T256 - 1) / T256, T256, 0, stream>>>(w2[0], wt2_0, HDIM, HDIM);
    k_wt_pad<<<(HDIM * HDIM + T256 - 1) / T256, T256, 0, stream>>>(w2[1], wt2_1, HDIM, HDIM);
    k_wt_pad<<<(HDIM * HDIM + T256 - 1) / T256, T256, 0, stream>>>(w2[2], wt2_2, HDIM, HDIM);

    // ---- layer 0 ----
    k_pad_copy<<<(NN * H0S + T256 - 1) / T256, T256, 0, stream>>>(feat, h0, NN, IN_DIM, H0S);
    k_scatter<<<sc_blocks, T256, 0, stream>>>(feat, h0, adj, EE, IN_DIM, H0S);
    k_mlp<H0S, H0K><<<mlp_blocks, 128, 0, stream>>>(h0,
                                          wt1_0, b1[0], wt2_0, b2[0],
                                          ga[0], be[0], mu[0], va[0],
                                          xA, pool, ib, 0, NN);

    // ---- layer 1 ----
    k_pad_copy<<<(NN * HDIM + T256 - 1) / T256, T256, 0, stream>>>(xA, hB, NN, HDIM, HDIM);
    k_scatter<<<sc_blocks, T256, 0, stream>>>(xA, hB, adj, EE, HDIM, HDIM);
    k_mlp<HDIM, HDIM><<<mlp_blocks, 128, 0, stream>>>(hB,
                                          wt1_1, b1[1], wt2_1, b2[1],
                                          ga[1], be[1], mu[1], va[1],
                                          xC, pool, ib, HDIM, NN);

    // ---- layer 2 ----
    k_pad_copy<<<(NN * HDIM + T256 - 1) / T256, T256, 0, stream>>>(xC, hB, NN, HDIM, HDIM);
    k_scatter<<<sc_blocks, T256, 0, stream>>>(xC, hB, adj, EE, HDIM, HDIM);
    k_mlp<HDIM, HDIM><<<mlp_blocks, 128, 0, stream>>>(hB,
                                          wt1_2, b1[2], wt2_2, b2[2],
                                          ga[2], be[2], mu[2], va[2],
                                          xA, pool, ib, 2 * HDIM, NN);

    // ---- finalize max pool -> output [G, 384] f32 ----
    k_pool_fin<<<(GG * POOLW + T256 - 1) / T256, T256, 0, stream>>>(pool, (float*)d_out, GG * POOLW);
}